// GAT_89704686944362
// MI455X (gfx1250) — compile-verified
//
#include <hip/hip_runtime.h>
#include <math.h>

typedef float v2f __attribute__((ext_vector_type(2)));
typedef float v4f __attribute__((ext_vector_type(4)));
typedef float v8f __attribute__((ext_vector_type(8)));

__device__ __forceinline__ float lrelu02(float v) { return v > 0.f ? v : 0.2f * v; }

// Float atomic max via monotonic int mapping: non-negative floats order as signed
// ints (global_atomic_max_i32); negative floats order reversed as unsigned
// (global_atomic_min_u32). Init value must be -inf (0xff800000).
__device__ __forceinline__ void atomicMaxF(float* addr, float v) {
  if (v >= 0.f) atomicMax((int*)addr, __float_as_int(v));
  else          atomicMin((unsigned int*)addr, __float_as_uint(v));
}

// ---------------------------------------------------------------------------
// H[M,NOUT] = X[M,K] @ W[K,NOUT], f32, via V_WMMA_F32_16X16X4_F32.
// Block = 256 threads = 8 waves; each wave computes a 32x64 output tile.
// The block's 64-column W strip is staged in LDS in WMMA B-FRAGMENT ORDER:
// K-row pairs interleaved, so fragment {W[2p][c], W[2p+1][c]} is one
// contiguous float2 -> a single ds_load_b64 into an even-aligned VGPR pair
// (no register shuffles before the WMMA). Pair-row stride 80 float2s puts the
// lane<16 / lane>=16 halves (delta = 1 pair-row = bank offset 32) on disjoint
// 32-bank halves. K and NOUT are template constants -> immediate offsets.
// A layout (ISA 7.12.2, 32-bit A 16x4): lane&15 -> M, lane>>4 selects K pair.
// C/D layout: VGPR r holds M=r (lanes 0-15) / M=r+8 (lanes 16-31), N=lane&15.
// ---------------------------------------------------------------------------
template <int K, int NOUT>
__global__ __launch_bounds__(256) void gat_gemm_wmma(
    const float* __restrict__ X, const float* __restrict__ W,
    float* __restrict__ H, int M) {
  constexpr int PSTR = 80;                    // pair-row stride in float2 units
  __shared__ float lw[(K / 2) * PSTR * 2];    // interleaved K-pair fragments

  const int col0 = blockIdx.y * 64;

  // ---- cooperative stage: rows 2p,2p+1 (b128 each) -> 4 interleaved b64 ----
  for (int j = threadIdx.x; j < (K / 2) * 16; j += 256) {
    int p  = j >> 4;
    int cc = (j & 15) << 2;
    v4f w0 = *(const v4f*)(W + (long)(2 * p + 0) * NOUT + col0 + cc);
    v4f w1 = *(const v4f*)(W + (long)(2 * p + 1) * NOUT + col0 + cc);
#pragma unroll
    for (int i = 0; i < 4; ++i) {
      v2f f; f.x = w0[i]; f.y = w1[i];
      *(v2f*)(&lw[(p * PSTR + cc + i) * 2]) = f;
    }
  }
  __syncthreads();

  const int lane = threadIdx.x & 31;
  const int wave = blockIdx.x * 8 + (threadIdx.x >> 5);
  const int row0 = wave * 32;
  if (row0 >= M) return;                      // wave-uniform: EXEC stays all-1s

  const int kph  = lane >> 4;                 // 0 -> K{0,1}, 1 -> K{2,3}
  const int ncol = lane & 15;
  const float* xr0 = X + (long)(row0 + (lane & 15)) * K;
  const float* xr1 = xr0 + (long)16 * K;

  v8f acc0[4] = {v8f{}, v8f{}, v8f{}, v8f{}};
  v8f acc1[4] = {v8f{}, v8f{}, v8f{}, v8f{}};

#pragma unroll
  for (int k = 0; k < K; k += 4) {
    const int kb = k + 2 * kph;               // A: two consecutive K columns
    const int pb = (k >> 1) + kph;            // B: fragment pair-row index
    v2f a0 = *(const v2f*)(xr0 + kb);
    v2f a1 = *(const v2f*)(xr1 + kb);
    v2f b[4];
#pragma unroll
    for (int t = 0; t < 4; ++t)
      b[t] = *(const v2f*)(&lw[(pb * PSTR + t * 16 + ncol) * 2]);
#pragma unroll
    for (int t = 0; t < 4; ++t) {
      acc0[t] = __builtin_amdgcn_wmma_f32_16x16x4_f32(
          false, a0, false, b[t], (short)0, acc0[t], false, false);
      acc1[t] = __builtin_amdgcn_wmma_f32_16x16x4_f32(
          false, a1, false, b[t], (short)0, acc1[t], false, false);
    }
  }

  float* hr0 = H + (long)(row0 + kph * 8) * NOUT + col0 + ncol;
  float* hr1 = hr0 + (long)16 * NOUT;
#pragma unroll
  for (int r = 0; r < 8; ++r) {
#pragma unroll
    for (int t = 0; t < 4; ++t) {
      hr0[(long)r * NOUT + t * 16] = acc0[t][r];
      hr1[(long)r * NOUT + t * 16] = acc1[t][r];
    }
  }
}

// al_s[n,h] = <h[n,h,:], a_s[h,:]> ; same for al_d. One thread per (node,head).
template <int C>
__global__ void attn_coef(const float* __restrict__ Hf,
                          const float* __restrict__ As, const float* __restrict__ Ad,
                          float* __restrict__ als, float* __restrict__ ald, int Nn) {
  int i = blockIdx.x * blockDim.x + threadIdx.x;
  if (i >= Nn * 8) return;
  int h = i & 7;
  const float* hp = Hf + (long)i * C;
  float ss = 0.f, dd = 0.f;
#pragma unroll
  for (int c = 0; c < C; ++c) {
    float v = hp[c];
    ss += v * As[h * C + c];
    dd += v * Ad[h * C + c];
  }
  als[i] = ss;
  ald[i] = dd;
}

__global__ void seg_init(float* __restrict__ mx, float* __restrict__ sm,
                         float* __restrict__ acc, int nH, int nAcc) {
  int i = blockIdx.x * blockDim.x + threadIdx.x;
  if (i < nAcc) acc[i] = 0.f;
  if (i < nH) { mx[i] = __int_as_float(0xff800000); sm[i] = 0.f; }
}

__global__ void edge_max(const int* __restrict__ src, const int* __restrict__ dst,
                         int E, int Nn, const float* __restrict__ als,
                         const float* __restrict__ ald, float* __restrict__ mx) {
  int e = blockIdx.x * blockDim.x + threadIdx.x;
  if (e >= E + Nn) return;
  int s = (e < E) ? src[e] : (e - E);
  int d = (e < E) ? dst[e] : (e - E);
#pragma unroll
  for (int h = 0; h < 8; ++h) {
    float v = lrelu02(als[s * 8 + h] + ald[d * 8 + h]);
    atomicMaxF(&mx[d * 8 + h], v);
  }
}

__global__ void edge_sum(const int* __restrict__ src, const int* __restrict__ dst,
                         int E, int Nn, const float* __restrict__ als,
                         const float* __restrict__ ald, const float* __restrict__ mx,
                         float* __restrict__ sm) {
  int e = blockIdx.x * blockDim.x + threadIdx.x;
  if (e >= E + Nn) return;
  int s = (e < E) ? src[e] : (e - E);
  int d = (e < E) ? dst[e] : (e - E);
#pragma unroll
  for (int h = 0; h < 8; ++h) {
    float v = lrelu02(als[s * 8 + h] + ald[d * 8 + h]);
    atomicAdd(&sm[d * 8 + h], __expf(v - mx[d * 8 + h]));
  }
}

// One thread per (edge, head); scatter alpha * h[src,h,:] into acc[dst,h,:].
template <int C>
__global__ void edge_aggr(const int* __restrict__ src, const int* __restrict__ dst,
                          int E, int Nn, const float* __restrict__ als,
                          const float* __restrict__ ald, const float* __restrict__ mx,
                          const float* __restrict__ sm, const float* __restrict__ Hf,
                          float* __restrict__ acc) {
  int i = blockIdx.x * blockDim.x + threadIdx.x;
  if (i >= (E + Nn) * 8) return;
  int e = i >> 3, h = i & 7;
  int s = (e < E) ? src[e] : (e - E);
  int d = (e < E) ? dst[e] : (e - E);
  float v = lrelu02(als[s * 8 + h] + ald[d * 8 + h]);
  float alpha = __expf(v - mx[d * 8 + h]) / (sm[d * 8 + h] + 1e-16f);
  const float* hp = Hf + ((long)s * 8 + h) * C;
  float* ap = acc + ((long)d * 8 + h) * C;
#pragma unroll
  for (int c = 0; c < C; ++c) atomicAdd(&ap[c], hp[c] * alpha);
}

// Mid layers: x_next = leaky_relu(acc + b), Fout fixed at 64 (mask not modulo).
__global__ void finalize_mid(const float* __restrict__ acc, const float* __restrict__ b,
                             float* __restrict__ xo, int Nn) {
  int i = blockIdx.x * blockDim.x + threadIdx.x;
  if (i >= Nn * 64) return;
  xo[i] = lrelu02(acc[i] + b[i & 63]);
}

// Last layer: head-mean + bias, then log_softmax over 40 classes. Thread/node.
__global__ void finalize_out(const float* __restrict__ acc, const float* __restrict__ b,
                             float* __restrict__ out, int Nn) {
  int n = blockIdx.x * blockDim.x + threadIdx.x;
  if (n >= Nn) return;
  const float* ap = acc + (long)n * 320;
  float y[40];
  float mmax = -3.402823466e38f;
#pragma unroll
  for (int c = 0; c < 40; ++c) {
    float t = 0.f;
#pragma unroll
    for (int h = 0; h < 8; ++h) t += ap[h * 40 + c];
    float v = t * 0.125f + b[c];
    y[c] = v;
    mmax = fmaxf(mmax, v);
  }
  float ssum = 0.f;
#pragma unroll
  for (int c = 0; c < 40; ++c) ssum += __expf(y[c] - mmax);
  float lse = __logf(ssum) + mmax;
#pragma unroll
  for (int c = 0; c < 40; ++c) out[(long)n * 40 + c] = y[c] - lse;
}

extern "C" void kernel_launch(void* const* d_in, const int* in_sizes, int n_in,
                              void* d_out, int out_size, void* d_ws, size_t ws_size,
                              hipStream_t stream) {
  const int Nn = in_sizes[0] / 128;   // 100000
  const int E  = in_sizes[1] / 2;     // 1600000
  const int Etot = E + Nn;            // with self-loops

  const float* x0 = (const float*)d_in[0];
  const int* ei   = (const int*)d_in[1];
  const int* src  = ei;
  const int* dst  = ei + E;

  char* ws = (char*)d_ws;
  size_t off = 0;
  auto carve = [&](size_t bytes) -> float* {
    float* p = (float*)(ws + off);
    off += (bytes + 255) & ~(size_t)255;
    return p;
  };
  float* h_buf = carve((size_t)Nn * 320 * sizeof(float));
  float* acc   = carve((size_t)Nn * 320 * sizeof(float));
  float* x_buf = carve((size_t)Nn * 64 * sizeof(float));
  float* als   = carve((size_t)Nn * 8 * sizeof(float));
  float* ald   = carve((size_t)Nn * 8 * sizeof(float));
  float* mx    = carve((size_t)Nn * 8 * sizeof(float));
  float* sm    = carve((size_t)Nn * 8 * sizeof(float));

  struct LayerDesc { int fin, C, wi; };
  const LayerDesc L[5] = { {128, 8, 2}, {64, 8, 6}, {64, 8, 10}, {64, 8, 14}, {64, 40, 18} };

  const int TB = 256;
  const int rowBlocks = (Nn + 255) / 256;   // 256 rows per block (8 waves x 32)
  const float* X = x0;
  for (int li = 0; li < 5; ++li) {
    const float* W  = (const float*)d_in[L[li].wi + 0];
    const float* As = (const float*)d_in[L[li].wi + 1];
    const float* Ad = (const float*)d_in[L[li].wi + 2];
    const float* B  = (const float*)d_in[L[li].wi + 3];
    const int C = L[li].C;
    const int Fout = 8 * C;

    // 1) dense projection h = X @ W (f32 WMMA, LDS-staged fragment-order B)
    if (li == 0) {
      gat_gemm_wmma<128, 64><<<dim3(rowBlocks, 1), TB, 0, stream>>>(X, W, h_buf, Nn);
    } else if (C == 8) {
      gat_gemm_wmma<64, 64><<<dim3(rowBlocks, 1), TB, 0, stream>>>(X, W, h_buf, Nn);
    } else {
      gat_gemm_wmma<64, 320><<<dim3(rowBlocks, 5), TB, 0, stream>>>(X, W, h_buf, Nn);
    }
    // 2) attention coefficients
    {
      int t = Nn * 8;
      if (C == 8)
        attn_coef<8><<<(t + TB - 1) / TB, TB, 0, stream>>>(h_buf, As, Ad, als, ald, Nn);
      else
        attn_coef<40><<<(t + TB - 1) / TB, TB, 0, stream>>>(h_buf, As, Ad, als, ald, Nn);
    }
    // 3) init segment max/sum + accumulator
    {
      int t = Nn * Fout;
      seg_init<<<(t + TB - 1) / TB, TB, 0, stream>>>(mx, sm, acc, Nn * 8, t);
    }
    // 4) per-dst max, 5) per-dst sum of exp, 6) weighted aggregation
    edge_max<<<(Etot + TB - 1) / TB, TB, 0, stream>>>(src, dst, E, Nn, als, ald, mx);
    edge_sum<<<(Etot + TB - 1) / TB, TB, 0, stream>>>(src, dst, E, Nn, als, ald, mx, sm);
    {
      int t = Etot * 8;
      if (C == 8)
        edge_aggr<8><<<(t + TB - 1) / TB, TB, 0, stream>>>(src, dst, E, Nn, als, ald, mx, sm, h_buf, acc);
      else
        edge_aggr<40><<<(t + TB - 1) / TB, TB, 0, stream>>>(src, dst, E, Nn, als, ald, mx, sm, h_buf, acc);
    }
    // 7) epilogue
    if (li < 4) {
      int t = Nn * 64;
      finalize_mid<<<(t + TB - 1) / TB, TB, 0, stream>>>(acc, B, x_buf, Nn);
      X = x_buf;
    } else {
      finalize_out<<<(Nn + TB - 1) / TB, TB, 0, stream>>>(acc, B, (float*)d_out, Nn);
    }
  }
}